// GravNetDGNLayer_20693152432520
// MI455X (gfx1250) — compile-verified
//
#include <hip/hip_runtime.h>
#include <hip/hip_bf16.h>
#include <math.h>

// Problem constants (match reference)
#define N_NODES   65536
#define B_GRAPHS  64
#define NPG       1024
#define KNN       16
#define IN_CH     64
#define OUT_CH    128
#define SPACE     4
#define PROP      64
#define AGG_CH    256      // 4*PROP
#define KTOT      320      // IN_CH + AGG_CH

typedef float v2f __attribute__((ext_vector_type(2)));
typedef float v8f __attribute__((ext_vector_type(8)));

__device__ __forceinline__ v8f wmma4(v2f a, v2f b, v8f c) {
  // V_WMMA_F32_16X16X4_F32 : D = A(16x4) * B(4x16) + C(16x16), fp32
  return __builtin_amdgcn_wmma_f32_16x16x4_f32(false, a, false, b, (short)0, c,
                                               false, false);
}

// ---------------------------------------------------------------------------
// Kernel 1: s = x @ Ws.T + bs   ([N,4]); tiny, VALU
// ---------------------------------------------------------------------------
__global__ void proj_s_kernel(const float* __restrict__ x,
                              const float* __restrict__ Ws,
                              const float* __restrict__ bs,
                              float* __restrict__ s) {
  __shared__ float w[SPACE * IN_CH];
  for (int t = threadIdx.x; t < SPACE * IN_CH; t += blockDim.x) w[t] = Ws[t];
  __syncthreads();
  int i = blockIdx.x * blockDim.x + threadIdx.x;
  const float* xr = x + (size_t)i * IN_CH;
  float a0 = bs[0], a1 = bs[1], a2 = bs[2], a3 = bs[3];
#pragma unroll 8
  for (int k = 0; k < IN_CH; ++k) {
    float xv = xr[k];
    a0 += xv * w[0 * IN_CH + k];
    a1 += xv * w[1 * IN_CH + k];
    a2 += xv * w[2 * IN_CH + k];
    a3 += xv * w[3 * IN_CH + k];
  }
  float4 r = make_float4(a0, a1, a2, a3);
  ((float4*)s)[i] = r;
}

// ---------------------------------------------------------------------------
// Kernel 2: generic fp32 WMMA GEMM  Y[N,C] = X[N,K] @ W[C,K].T + bias
// 256 threads = 8 waves; wave -> one 16x16 tile; blockIdx.x -> col tile,
// blockIdx.y -> group of 8 row tiles. W strip cached in LDS.
// ---------------------------------------------------------------------------
template <int K, int C>
__global__ void gemm_wmma_kernel(const float* __restrict__ X,
                                 const float* __restrict__ W,
                                 const float* __restrict__ bias,
                                 float* __restrict__ Y) {
  constexpr int STRIDE = K + ((4 - (K % 64) + 64) % 64);  // stride % 64 == 4
  __shared__ float lw[16 * STRIDE];
  const int col0 = blockIdx.x * 16;
  for (int t = threadIdx.x; t < 16 * K; t += 256) {
    int cl = t / K, kk = t - cl * K;
    lw[cl * STRIDE + kk] = W[(size_t)(col0 + cl) * K + kk];
  }
  __syncthreads();

  const int lane = threadIdx.x & 31;
  const int wave = threadIdx.x >> 5;
  const int m = lane & 15;        // A row / B col within tile
  const int hb = lane >> 4;       // lane half selects K pair
  const int row0 = (blockIdx.y * 8 + wave) * 16;

  const float* xrow = X + (size_t)(row0 + m) * K + 2 * hb;
  const float* brow = &lw[m * STRIDE + 2 * hb];

  v8f acc = {0.f, 0.f, 0.f, 0.f, 0.f, 0.f, 0.f, 0.f};
#pragma unroll 4
  for (int k0 = 0; k0 < K; k0 += 4) {
    v2f a = *(const v2f*)(xrow + k0);
    v2f b = *(const v2f*)(brow + k0);
    acc = wmma4(a, b, acc);
  }
  const float bv = bias[col0 + m];
#pragma unroll
  for (int r = 0; r < 8; ++r)
    Y[(size_t)(row0 + r + 8 * hb) * C + col0 + m] = acc[r] + bv;
}

// ---------------------------------------------------------------------------
// Kernel 3: exact kNN per graph (1024 nodes, 4-D space, K=16, no self)
// One block (1024 threads) per graph; s tile in LDS; register top-16.
// ---------------------------------------------------------------------------
__global__ void knn_kernel(const float* __restrict__ s, int* __restrict__ idx) {
  __shared__ float4 ls[NPG];
  const int b = blockIdx.x;
  const int i = threadIdx.x;
  ls[i] = ((const float4*)s)[b * NPG + i];
  __syncthreads();
  const float4 si = ls[i];

  float bd[KNN];
  int bi[KNN];
#pragma unroll
  for (int t = 0; t < KNN; ++t) { bd[t] = 3.0e38f; bi[t] = 0; }

  for (int j = 0; j < NPG; ++j) {
    if (j == i) continue;
    float4 sj = ls[j];
    float dx = si.x - sj.x, dy = si.y - sj.y, dz = si.z - sj.z, dw = si.w - sj.w;
    float d2 = dx * dx + dy * dy + dz * dz + dw * dw;
    if (d2 < bd[KNN - 1]) {
      float dv = d2; int jv = j;
#pragma unroll
      for (int t = 0; t < KNN; ++t) {       // bubble-in, static indices only
        if (dv < bd[t]) {
          float td = bd[t]; int ti = bi[t];
          bd[t] = dv; bi[t] = jv;
          dv = td; jv = ti;
        }
      }
    }
  }
  int* ip = idx + (size_t)(b * NPG + i) * KNN;
#pragma unroll
  for (int t = 0; t < KNN; ++t) ip[t] = bi[t];
}

// ---------------------------------------------------------------------------
// Kernel 4: per-node aggregation -> agg [N, 256] = [mean | max | smooth | deriv]
// One wave per node (8 waves / 256-thread block). Lanes 0..15 own neighbors,
// all 32 lanes own channels {lane, lane+32}.
// ---------------------------------------------------------------------------
__global__ void agg_kernel(const float* __restrict__ s,
                           const float* __restrict__ h,
                           const float* __restrict__ z,
                           const int* __restrict__ idx,
                           float* __restrict__ agg) {
  const int lane = threadIdx.x & 31;
  const int wave = threadIdx.x >> 5;
  const int i = blockIdx.x * 8 + wave;
  const int b = i >> 10;                 // graph id (nodes grouped by 1024)
  (void)b;

  const int k = lane & 15;
  const int j = idx[(size_t)i * KNN + k];

  const float4 si = ((const float4*)s)[i];
  const float4 sj = ((const float4*)s)[j];
  float dx = si.x - sj.x, dy = si.y - sj.y, dzc = si.z - sj.z, dw = si.w - sj.w;
  float d2 = dx * dx + dy * dy + dzc * dzc + dw * dw;
  float ew = expf(-10.0f * d2);

  float dz = z[j] - z[i];
  float asum = fabsf(dz);
  asum += __shfl_xor(asum, 1, 32);
  asum += __shfl_xor(asum, 2, 32);
  asum += __shfl_xor(asum, 4, 32);
  asum += __shfl_xor(asum, 8, 32);       // sum over the 16-lane neighbor group
  float F = dz / (asum + 1e-8f);

  const int c0 = lane;                   // channels c0 and c0+32
  float amean0 = 0.f, amean1 = 0.f;
  float amax0 = -3.0e38f, amax1 = -3.0e38f;
  float asm0 = 0.f, asm1 = 0.f;
  float adv0 = 0.f, adv1 = 0.f;

#pragma unroll
  for (int kk = 0; kk < KNN; ++kk) {
    int jk = __shfl(j, kk, 32);
    float ewk = __shfl(ew, kk, 32);
    float Fk = __shfl(F, kk, 32);
    float aFk = fabsf(Fk);
    const float* hr = h + (size_t)jk * PROP;
    float m0 = hr[c0] * ewk;
    float m1 = hr[c0 + 32] * ewk;
    amean0 += m0;            amean1 += m1;
    amax0 = fmaxf(amax0, m0); amax1 = fmaxf(amax1, m1);
    asm0 += m0 * aFk;        asm1 += m1 * aFk;
    adv0 += m0 * Fk;         adv1 += m1 * Fk;
  }

  float* ar = agg + (size_t)i * AGG_CH;
  const float inv = 1.0f / (float)KNN;
  ar[c0] = amean0 * inv;        ar[c0 + 32] = amean1 * inv;
  ar[64 + c0] = amax0;          ar[64 + c0 + 32] = amax1;
  ar[128 + c0] = asm0 * inv;    ar[128 + c0 + 32] = asm1 * inv;
  ar[192 + c0] = adv0;          ar[192 + c0 + 32] = adv1;
}

// ---------------------------------------------------------------------------
// Kernel 5: out = x @ Wskip.T + agg @ Wagg.T + bagg   (K split 64 + 256, WMMA)
// ---------------------------------------------------------------------------
__global__ void out_gemm_kernel(const float* __restrict__ x,
                                const float* __restrict__ agg,
                                const float* __restrict__ Wskip,
                                const float* __restrict__ Wagg,
                                const float* __restrict__ bagg,
                                float* __restrict__ out) {
  constexpr int STRIDE = 324;            // 324 % 64 == 4 -> conflict-free b64
  __shared__ float lw[16 * STRIDE];
  const int col0 = blockIdx.x * 16;
  for (int t = threadIdx.x; t < 16 * KTOT; t += 256) {
    int cl = t / KTOT, kk = t - cl * KTOT;
    float wv = (kk < IN_CH) ? Wskip[(size_t)(col0 + cl) * IN_CH + kk]
                            : Wagg[(size_t)(col0 + cl) * AGG_CH + (kk - IN_CH)];
    lw[cl * STRIDE + kk] = wv;
  }
  __syncthreads();

  const int lane = threadIdx.x & 31;
  const int wave = threadIdx.x >> 5;
  const int m = lane & 15;
  const int hb = lane >> 4;
  const int row0 = (blockIdx.y * 8 + wave) * 16;

  const float* xrow = x + (size_t)(row0 + m) * IN_CH + 2 * hb;
  const float* arow = agg + (size_t)(row0 + m) * AGG_CH + 2 * hb;
  const float* brow = &lw[m * STRIDE + 2 * hb];

  v8f acc = {0.f, 0.f, 0.f, 0.f, 0.f, 0.f, 0.f, 0.f};
#pragma unroll 4
  for (int k0 = 0; k0 < IN_CH; k0 += 4) {
    v2f a = *(const v2f*)(xrow + k0);
    v2f b = *(const v2f*)(brow + k0);
    acc = wmma4(a, b, acc);
  }
#pragma unroll 4
  for (int k0 = 0; k0 < AGG_CH; k0 += 4) {
    v2f a = *(const v2f*)(arow + k0);
    v2f b = *(const v2f*)(brow + IN_CH + k0);
    acc = wmma4(a, b, acc);
  }
  const float bv = bagg[col0 + m];
#pragma unroll
  for (int r = 0; r < 8; ++r)
    out[(size_t)(row0 + r + 8 * hb) * OUT_CH + col0 + m] = acc[r] + bv;
}

// ---------------------------------------------------------------------------
// Kernels 6-8: deterministic BatchNorm (training stats, biased var) + ReLU
// ---------------------------------------------------------------------------
__global__ void bn_partial_kernel(const float* __restrict__ out,
                                  float* __restrict__ psum,
                                  float* __restrict__ psq) {
  const int c = threadIdx.x;             // 128 threads = channels
  const int g = blockIdx.x;              // 256 row groups of 256 rows
  float s = 0.f, q = 0.f;
  const int r0 = g * 256;
  for (int r = r0; r < r0 + 256; ++r) {
    float v = out[(size_t)r * OUT_CH + c];
    s += v;
    q += v * v;
  }
  psum[g * OUT_CH + c] = s;
  psq[g * OUT_CH + c] = q;
}

__global__ void bn_finalize_kernel(const float* __restrict__ psum,
                                   const float* __restrict__ psq,
                                   float* __restrict__ stats) {
  const int c = threadIdx.x;             // 128 threads
  float s = 0.f, q = 0.f;
  for (int g = 0; g < 256; ++g) {        // fixed order -> deterministic
    s += psum[g * OUT_CH + c];
    q += psq[g * OUT_CH + c];
  }
  const float invN = 1.0f / (float)N_NODES;
  float mu = s * invN;
  float var = q * invN - mu * mu;
  stats[c] = mu;
  stats[OUT_CH + c] = rsqrtf(var + 1e-5f);
}

__global__ void bn_apply_kernel(float* __restrict__ out,
                                const float* __restrict__ stats,
                                const float* __restrict__ gamma,
                                const float* __restrict__ beta) {
  size_t t = (size_t)blockIdx.x * 256 + threadIdx.x;
  int c = (int)(t & (OUT_CH - 1));
  float v = out[t];
  v = gamma[c] * (v - stats[c]) * stats[OUT_CH + c] + beta[c];
  out[t] = fmaxf(v, 0.0f);
}

// ---------------------------------------------------------------------------
// Launch
// ---------------------------------------------------------------------------
extern "C" void kernel_launch(void* const* d_in, const int* in_sizes, int n_in,
                              void* d_out, int out_size, void* d_ws, size_t ws_size,
                              hipStream_t stream) {
  const float* x     = (const float*)d_in[0];
  const float* z     = (const float*)d_in[1];
  // d_in[2] = batch (int32) — graphs are exactly contiguous blocks of 1024, unused
  const float* Ws    = (const float*)d_in[3];
  const float* bs    = (const float*)d_in[4];
  const float* Wh    = (const float*)d_in[5];
  const float* bh    = (const float*)d_in[6];
  const float* Wskip = (const float*)d_in[7];
  const float* Wagg  = (const float*)d_in[8];
  const float* bagg  = (const float*)d_in[9];
  const float* gamma = (const float*)d_in[10];
  const float* beta  = (const float*)d_in[11];
  float* out = (float*)d_out;

  // Workspace carving (byte offsets, all 256-aligned)
  char* ws = (char*)d_ws;
  float* s_buf   = (float*)(ws + 0);                    //  N*4   f32 (1 MB)
  float* h_buf   = (float*)(ws + 1048576);              //  N*64  f32 (16 MB)
  int*   idx_buf = (int*)  (ws + 17825792);             //  N*16  i32 (4 MB)
  float* agg_buf = (float*)(ws + 22020096);             //  N*256 f32 (64 MB)
  float* psum    = (float*)(ws + 89128960);             //  256*128 f32
  float* psq     = (float*)(ws + 89260032);             //  256*128 f32
  float* stats   = (float*)(ws + 89391104);             //  2*128 f32

  // 1) s projection
  proj_s_kernel<<<N_NODES / 256, 256, 0, stream>>>(x, Ws, bs, s_buf);

  // 2) h projection: [N,64] = x @ Wh.T + bh  (fp32 WMMA)
  gemm_wmma_kernel<IN_CH, PROP>
      <<<dim3(PROP / 16, N_NODES / 128), 256, 0, stream>>>(x, Wh, bh, h_buf);

  // 3) exact kNN per graph
  knn_kernel<<<B_GRAPHS, NPG, 0, stream>>>(s_buf, idx_buf);

  // 4) edge weights + DGN aggregations
  agg_kernel<<<N_NODES / 8, 256, 0, stream>>>(s_buf, h_buf, z, idx_buf, agg_buf);

  // 5) fused output GEMM (skip + agg paths, fp32 WMMA)
  out_gemm_kernel<<<dim3(OUT_CH / 16, N_NODES / 128), 256, 0, stream>>>(
      x, agg_buf, Wskip, Wagg, bagg, out);

  // 6-8) BatchNorm (deterministic two-stage reduction) + ReLU
  bn_partial_kernel<<<256, OUT_CH, 0, stream>>>(out, psum, psq);
  bn_finalize_kernel<<<1, OUT_CH, 0, stream>>>(psum, psq, stats);
  bn_apply_kernel<<<(N_NODES * OUT_CH) / 256, 256, 0, stream>>>(out, stats, gamma, beta);
}